// _ProbeFold_77206332112991
// MI455X (gfx1250) — compile-verified
//
#include <hip/hip_runtime.h>
#include <cstdint>
#include <cstddef>

typedef float v4f __attribute__((ext_vector_type(4)));

constexpr int B = 4, P = 8, S = 2048, D = 2048;
constexpr int SD = S * D;                        // floats per (b,p) slab
constexpr int THREADS = 1024;
constexpr int VEC_PER_THREAD = 4;                // float4 per thread
constexpr int VEC_PER_BLOCK = THREADS * VEC_PER_THREAD;   // 4096 vec4 = 16384 floats
constexpr int BLOCKS_PER_BATCH = (SD / 4) / VEC_PER_BLOCK; // 256
constexpr int GRID = B * BLOCKS_PER_BATCH;       // 1024

__global__ __launch_bounds__(THREADS)
void probefold_kernel(const float* __restrict__ probes,
                      const float* __restrict__ scores,
                      const float* __restrict__ re_expand,
                      float* __restrict__ out) {
    __shared__ __align__(16) float s_re[P * D];  // 64 KiB, full re_expand

    const int t     = threadIdx.x;
    const int blk   = blockIdx.x;
    const int batch = blk >> 8;                  // / BLOCKS_PER_BATCH
    const int bblk  = blk & (BLOCKS_PER_BATCH - 1);

    // ---- Stage re_expand (64 KiB) into LDS via CDNA5 async global->LDS DMA ----
    {
        uint32_t lds_base = (uint32_t)(uintptr_t)(&s_re[0]);
        const char* gbase = (const char*)re_expand;
        #pragma unroll
        for (int j = 0; j < (P * D * 4) / (THREADS * 16); ++j) {   // 4 sweeps
            uint32_t off = (uint32_t)(t + j * THREADS) * 16u;
            uint32_t lp  = lds_base + off;
            const char* gp = gbase + off;
            asm volatile("global_load_async_to_lds_b128 %0, %1, off"
                         :: "v"(lp), "v"(gp) : "memory");
        }
        asm volatile("s_wait_asynccnt 0" ::: "memory");
    }

    // ---- Per-batch top-2 + softmax (uniform addresses -> scalar loads) ----
    float best = scores[batch * P]; int bi = 0;
    float sec  = -3.4e38f;          int si = 0;
    #pragma unroll
    for (int p = 1; p < P; ++p) {
        float v = scores[batch * P + p];
        if (v > best)     { sec = best; si = bi; best = v; bi = p; }
        else if (v > sec) { sec = v;    si = p; }
    }
    float e  = expf(sec - best);
    float wb = 1.0f / (1.0f + e);   // weight of best
    float ws = e * wb;              // weight of second

    __syncthreads();                // LDS staging visible to whole workgroup

    // Each thread's d position is invariant across its 4 iterations:
    // f = 16384*bblk + 4096*i + 4*t  =>  f mod 2048 == (4*t) mod 2048
    const int d = (t * 4) & (D - 1);
    v4f rre[P];
    #pragma unroll
    for (int p = 0; p < P; ++p)
        rre[p] = *(const v4f*)(&s_re[p * D + d]);

    const float* pb = probes + ((size_t)batch * P + bi) * SD;
    const float* ps = probes + ((size_t)batch * P + si) * SD;
    float*       ob = out    +  (size_t)batch * P * SD;

    #pragma unroll
    for (int i = 0; i < VEC_PER_THREAD; ++i) {
        const int f = (bblk * VEC_PER_BLOCK + i * THREADS + t) * 4;
        v4f a = __builtin_nontemporal_load((const v4f*)(pb + f));
        v4f b = __builtin_nontemporal_load((const v4f*)(ps + f));
        v4f m = a * wb + b * ws;
        #pragma unroll
        for (int p = 0; p < P; ++p) {
            v4f r = m + rre[p];
            __builtin_nontemporal_store(r, (v4f*)(ob + (size_t)p * SD + f));
        }
    }
}

extern "C" void kernel_launch(void* const* d_in, const int* in_sizes, int n_in,
                              void* d_out, int out_size, void* d_ws, size_t ws_size,
                              hipStream_t stream) {
    (void)in_sizes; (void)n_in; (void)out_size; (void)d_ws; (void)ws_size;
    const float* probes    = (const float*)d_in[0];
    const float* scores    = (const float*)d_in[1];
    const float* re_expand = (const float*)d_in[2];
    // d_in[3] is top_k == 2 (hardcoded in the kernel's top-2 selection)
    float* out = (float*)d_out;
    probefold_kernel<<<GRID, THREADS, 0, stream>>>(probes, scores, re_expand, out);
}